// ModelNew_25056839205270
// MI455X (gfx1250) — compile-verified
//
#include <hip/hip_runtime.h>
#include <stdint.h>

// ---------------------------------------------------------------------------
// SA-Net shuffle attention, fused single-pass kernel for MI455X (gfx1250).
//
// Shapes (hardcoded from reference): B=64, C=512, H=W=28, G=8
//   -> 512 groups of (Cg=64 channels x HW=784), group tile = 196KB fp32.
//
// Roofline: ~98 MB read + ~98 MB write vs ~0.5 GFLOP -> pure bandwidth,
// ~8.4us floor at 23.3 TB/s. CDNA5 levers used:
//   * 320KB LDS/WGP: whole (b,g) tile resides in LDS -> single HBM read
//     (no second stats pass over global memory).
//   * GLOBAL_LOAD_ASYNC_TO_LDS_B128 (ASYNCcnt) bulk copy, no VGPR round-trip.
// All loops have compile-time trip counts (fully unrolled + one predicated
// tail) so no exec-masked loop bookkeeping around the async issues.
// ---------------------------------------------------------------------------

#define BLOCK        1024              // 32 wave32 waves
#define G_           8
#define CG           64                // channels per group
#define CHALF        32                // channels per half-group
#define HW           784               // 28*28
#define VEC4_PER_ROW 196               // 784/4
#define ELEMS        (CG * HW)         // 50176 floats per group tile
#define VEC4S        (ELEMS / 4)       // 12544 float4 per tile = 12*1024+256
#define X1_ELEMS     (CHALF * HW)      // 25088 = 24*1024+512
#define NCHAN        512
#define EPS          1e-5f

__device__ __forceinline__ float warp_sum(float v) {
#pragma unroll
  for (int m = 16; m; m >>= 1) v += __shfl_xor(v, m, 32);
  return v;
}

__device__ __forceinline__ float fast_sigmoid(float t) {
  return 1.0f / (1.0f + __expf(-t));
}

// Async copy of 16 bytes from global memory directly into LDS (no VGPR
// round-trip). CDNA5-only path, tracked with ASYNCcnt. Inline asm because the
// clang builtins for the async/TDM family differ in arity across ROCm 7.2 and
// the amdgpu-toolchain lane; the ISA mnemonic is stable.
__device__ __forceinline__ void async_ld_b128(uint32_t lds_byte_addr,
                                              const void* gptr) {
  asm volatile("global_load_async_to_lds_b128 %0, %1, off"
               :
               : "v"(lds_byte_addr), "v"((unsigned long long)(uintptr_t)gptr)
               : "memory");
}

__device__ __forceinline__ void wait_async_zero() {
  asm volatile("s_wait_asynccnt 0" ::: "memory");
}

extern "C" __global__ void __launch_bounds__(BLOCK)
sa_shuffle_attn_kernel(const float* __restrict__ x,
                       const float* __restrict__ cweight,
                       const float* __restrict__ cbias,
                       const float* __restrict__ sweight,
                       const float* __restrict__ sbias,
                       float* __restrict__ out) {
  // Dynamic LDS: tile (50176) + a0 gates (32) + reduce scratch (64) + stats(2)
  extern __shared__ float smem[];
  float* tile    = smem;                 // [CG][HW]
  float* s_a0    = smem + ELEMS;         // [CHALF] channel-attention gates
  float* s_red   = s_a0 + CHALF;         // [64]: 32 sums + 32 sumsq partials
  float* s_stats = s_red + 64;           // [2]: mu, rstd

  const int bg   = blockIdx.x;           // 0..511  (= b*G + g)
  const int b    = bg >> 3;
  const int g    = bg & (G_ - 1);
  const int tid  = threadIdx.x;
  const int wid  = tid >> 5;
  const int lane = tid & 31;

  // Group tile is contiguous in x: offset = bg * CG * HW
  const float* gbase = x + (size_t)bg * ELEMS;

  // ---- Phase 0: async bulk copy of the whole group tile into LDS ----------
  // Generic->LDS: low 32 bits of a flat shared-space pointer are the LDS
  // byte offset (ISA 10.2 aperture mapping).
  const uint32_t lbase = (uint32_t)(uintptr_t)tile;
#pragma unroll
  for (int k = 0; k < VEC4S / BLOCK; ++k) {          // 12 uniform issues
    const int i = tid + k * BLOCK;
    async_ld_b128(lbase + (uint32_t)(i * 16), (const void*)(gbase + i * 4));
  }
  if (tid < (VEC4S - (VEC4S / BLOCK) * BLOCK)) {     // 256-thread tail
    const int i = tid + (VEC4S / BLOCK) * BLOCK;
    async_ld_b128(lbase + (uint32_t)(i * 16), (const void*)(gbase + i * 4));
  }
  wait_async_zero();     // each wave drains its own ASYNCcnt
  __syncthreads();       // tile visible to the whole workgroup

  // ---- Phase 1: channel-attention gates for x0 rows (cg = 0..31) ----------
  // Wave w owns row w: lane-strided sum over 784 elements + shuffle tree.
  {
    const float* row = tile + wid * HW;  // wid in 0..31 == CHALF rows
    float s = 0.0f;
#pragma unroll
    for (int k = 0; k < HW / 32; ++k) s += row[lane + k * 32];  // 24 loads
    if (lane < (HW - (HW / 32) * 32)) s += row[lane + (HW / 32) * 32]; // 16
    s = warp_sum(s);
    if (lane == 0) {
      const float m = s * (1.0f / (float)HW);
      s_a0[wid] = fast_sigmoid(cweight[wid] * m + cbias[wid]);
    }
  }

  // ---- Phase 2: group mean/var over x1 (rows 32..63, 25088 elems) ---------
  {
    const float* x1t = tile + CHALF * HW;
    float s = 0.0f, ss = 0.0f;
#pragma unroll
    for (int k = 0; k < X1_ELEMS / BLOCK; ++k) {     // 24 uniform loads
      const float v = x1t[tid + k * BLOCK];
      s += v;
      ss += v * v;
    }
    if (tid < (X1_ELEMS - (X1_ELEMS / BLOCK) * BLOCK)) {  // 512-thread tail
      const float v = x1t[tid + (X1_ELEMS / BLOCK) * BLOCK];
      s += v;
      ss += v * v;
    }
    s  = warp_sum(s);
    ss = warp_sum(ss);
    if (lane == 0) {
      s_red[wid]      = s;
      s_red[32 + wid] = ss;
    }
  }
  __syncthreads();
  if (wid == 0) {
    float ps  = s_red[lane];
    float pss = s_red[32 + lane];
    ps  = warp_sum(ps);
    pss = warp_sum(pss);
    if (lane == 0) {
      const float mu  = ps * (1.0f / (float)X1_ELEMS);
      float var       = pss * (1.0f / (float)X1_ELEMS) - mu * mu;
      var             = fmaxf(var, 0.0f);
      s_stats[0]      = mu;
      s_stats[1]      = rsqrtf(var + EPS);
    }
  }
  __syncthreads();

  // ---- Phase 3: apply gates + fused 2-way channel shuffle, float4 stores --
  // y_cat channel cc = g*64 + cg ; shuffle: oc = 2*((g%4)*64 + cg) + (g/4)
  const float  mu    = s_stats[0];
  const float  rstd  = s_stats[1];
  const size_t obase = (size_t)b * NCHAN * HW;
  const int    gm4   = (g & 3) * CG;
  const int    ghi   = g >> 2;

#pragma unroll
  for (int k = 0; k < (VEC4S / BLOCK) + 1; ++k) {    // 12 uniform + 1 tail
    const int i = tid + k * BLOCK;
    if (k == VEC4S / BLOCK && i >= VEC4S) break;     // tail predicate
    const int row = i / VEC4_PER_ROW;                // cg in 0..63
    const int col = i - row * VEC4_PER_ROW;          // float4 column
    const float4 v = *(const float4*)(tile + row * HW + col * 4);
    float4 r;
    if (row < CHALF) {
      const float a = s_a0[row];
      r.x = v.x * a; r.y = v.y * a; r.z = v.z * a; r.w = v.w * a;
    } else {
      const int   c  = row - CHALF;
      const float sw = sweight[c];
      const float sb = sbias[c];
      r.x = v.x * fast_sigmoid(sw * ((v.x - mu) * rstd) + sb);
      r.y = v.y * fast_sigmoid(sw * ((v.y - mu) * rstd) + sb);
      r.z = v.z * fast_sigmoid(sw * ((v.z - mu) * rstd) + sb);
      r.w = v.w * fast_sigmoid(sw * ((v.w - mu) * rstd) + sb);
    }
    const int oc = 2 * (gm4 + row) + ghi;
    *(float4*)(out + obase + (size_t)oc * HW + col * 4) = r;
  }
}

extern "C" void kernel_launch(void* const* d_in, const int* in_sizes, int n_in,
                              void* d_out, int out_size, void* d_ws,
                              size_t ws_size, hipStream_t stream) {
  (void)in_sizes; (void)n_in; (void)d_ws; (void)ws_size; (void)out_size;
  const float* x  = (const float*)d_in[0];
  const float* cw = (const float*)d_in[1];
  const float* cb = (const float*)d_in[2];
  const float* sw = (const float*)d_in[3];
  const float* sb = (const float*)d_in[4];
  // d_in[5] is G (==8), hardcoded above.
  float* out = (float*)d_out;

  const size_t shmem = (size_t)(ELEMS + CHALF + 64 + 2) * sizeof(float); // ~196.5 KB
  dim3 grid(64 * G_);   // 512 groups
  dim3 block(BLOCK);
  hipLaunchKernelGGL(sa_shuffle_attn_kernel, grid, block, shmem, stream,
                     x, cw, cb, sw, sb, out);
}